// TrustRegionForwardWithUncertainty_70360154243611
// MI455X (gfx1250) — compile-verified
//
#include <hip/hip_runtime.h>
#include <math.h>

#define B_ 8
#define C_ 32
#define H_ 120
#define W_ 160
#define HW_ (H_ * W_)
#define CHW_ (C_ * HW_)
#define OUT_STRIDE (12 + CHW_)
#define MAX_ITER_ 3
#define DAMP_ 1e-4f
#define OCC_THRES_ 0.1f

typedef __attribute__((ext_vector_type(2))) float v2f;
typedef __attribute__((ext_vector_type(8))) float v8f;

// Workspace layout (float offsets)
#define WS_GX 0
#define WS_GY (WS_GX + B_ * CHW_)
#define WS_RT (WS_GY + B_ * CHW_)      // per batch: R(9) + t(3)
#define WS_PART (WS_RT + B_ * 12)      // per batch: 8x8 padded JtJ/rhs partials

// ---------------------------------------------------------------------------
// gfx1250 async global->LDS copy (ASYNCcnt path), saddr addressing mode.
// ---------------------------------------------------------------------------
__device__ __forceinline__ void async_load_f0(unsigned long long base,
                                              unsigned byteOff,
                                              unsigned ldsAddr) {
  asm volatile("global_load_async_to_lds_b32 %0, %1, %2"
               :: "v"(ldsAddr), "v"(byteOff), "s"(base)
               : "memory");
}

// ---------------------------------------------------------------------------
// Copy initial pose into workspace state; zero reduction partials.
// ---------------------------------------------------------------------------
__global__ void init_state_kernel(const float* __restrict__ Rin,
                                  const float* __restrict__ tin,
                                  float* __restrict__ ws) {
  int tid = threadIdx.x;  // 512 threads
  if (tid < B_ * 12) {
    int b = tid / 12, k = tid % 12;
    ws[WS_RT + tid] = (k < 9) ? Rin[b * 9 + k] : tin[b * 3 + (k - 9)];
  }
  if (tid < B_ * 64) ws[WS_PART + tid] = 0.0f;
}

// ---------------------------------------------------------------------------
// Sobel gradient of F1 (edge-replicated), normalized by magnitude. Done once.
// ---------------------------------------------------------------------------
__global__ void sobel_kernel(const float* __restrict__ F1, float* __restrict__ ws) {
  long idx = (long)blockIdx.x * 256 + threadIdx.x;  // over B*C*H*W (exact)
  int x = (int)(idx % W_);
  int y = (int)((idx / W_) % H_);
  long plane = idx / HW_;  // b*C + c
  const float* F = F1 + plane * HW_;
  int xm = max(x - 1, 0), xp = min(x + 1, W_ - 1);
  int ym = max(y - 1, 0), yp = min(y + 1, H_ - 1);
  float dx = (F[ym * W_ + xp] - F[ym * W_ + xm]) +
             2.0f * (F[y * W_ + xp] - F[y * W_ + xm]) +
             (F[yp * W_ + xp] - F[yp * W_ + xm]);
  float dy = (F[yp * W_ + xm] - F[ym * W_ + xm]) +
             2.0f * (F[yp * W_ + x] - F[ym * W_ + x]) +
             (F[yp * W_ + xp] - F[ym * W_ + xp]);
  float mag = sqrtf(dx * dx + dy * dy + 1e-8f);
  ws[WS_GX + idx] = dx / mag;
  ws[WS_GY + idx] = dy / mag;
}

// ---------------------------------------------------------------------------
// Main IRLS pass: per-pixel warp + bilinear gathers + Jacobian rows,
// normal-equation reduction via V_WMMA_F32_16X16X4_F32 (D = EᵀE, fp32).
// One lane = one pixel; 32 channels looped; 32 rows/wave per channel step.
// F0 (the one dense stream) is double-buffered into LDS with
// GLOBAL_LOAD_ASYNC_TO_LDS_B32, overlapped with the gather-bound body.
// ---------------------------------------------------------------------------
__global__ __launch_bounds__(256, 2) void irls_kernel(
    const float* __restrict__ F0, const float* __restrict__ F1,
    const float* __restrict__ invD0, const float* __restrict__ invD1,
    const float* __restrict__ Kin, float* __restrict__ ws,
    float* __restrict__ out, int write_weights) {
  __shared__ float rowbuf[8][32][8];  // per-wave 32 rows x 8 (7 used + zero pad)
  __shared__ float f0buf[2][256];     // async double buffer for F0 stream

  const int tid = threadIdx.x;
  const int wave = tid >> 5;
  const int lane = tid & 31;
  const int col = lane & 15;
  const int hlf = lane >> 4;
  const int b = blockIdx.x / (HW_ / 256);
  const int tile = blockIdx.x % (HW_ / 256);
  const int p = tile * 256 + tid;  // pixel id, exact coverage
  const int x = p % W_;
  const int y = p / W_;

  const float fx = Kin[b * 4 + 0], fy = Kin[b * 4 + 1];
  const float cx = Kin[b * 4 + 2], cy = Kin[b * 4 + 3];
  const float* Rt = ws + WS_RT + b * 12;
  const float R00 = Rt[0], R01 = Rt[1], R02 = Rt[2];
  const float R10 = Rt[3], R11 = Rt[4], R12 = Rt[5];
  const float R20 = Rt[6], R21 = Rt[7], R22 = Rt[8];
  const float t0 = Rt[9], t1 = Rt[10], t2 = Rt[11];

  const float px = ((float)x - cx) / fx;
  const float py = ((float)y - cy) / fy;
  const float id0 = invD0[b * HW_ + p];

  // Warp inverse depth: R*(px,py,1) + t*invD0
  const float sx = R00 * px + R01 * py + R02 + t0 * id0;
  const float sy = R10 * px + R11 * py + R12 + t1 * id0;
  const float sz = R20 * px + R21 * py + R22 + t2 * id0;
  const float u = sx / sz * fx + cx;
  const float v = sy / sz * fy + cy;
  const float inv_z = id0 / sz;
  const bool inview = (u > 0.0f) && (u < (float)(W_ - 1)) &&
                      (v > 0.0f) && (v < (float)(H_ - 1));

  // Bilinear sample setup (clamped like reference)
  const float uc = fminf(fmaxf(u, 0.0f), (float)(W_ - 1));
  const float vc = fminf(fmaxf(v, 0.0f), (float)(H_ - 1));
  const float x0f = floorf(uc), y0f = floorf(vc);
  const float fu = uc - x0f, fv = vc - y0f;
  const int x0 = (int)x0f, y0 = (int)y0f;
  const int x1 = min(x0 + 1, W_ - 1), y1 = min(y0 + 1, H_ - 1);
  const int o00 = y0 * W_ + x0, o01 = y0 * W_ + x1;
  const int o10 = y1 * W_ + x0, o11 = y1 * W_ + x1;
  const float w00 = (1.0f - fu) * (1.0f - fv), w01 = fu * (1.0f - fv);
  const float w10 = (1.0f - fu) * fv, w11 = fu * fv;

  const float* d1 = invD1 + (long)b * HW_;
  const float invD1w = d1[o00] * w00 + d1[o01] * w01 + d1[o10] * w10 + d1[o11] * w11;
  const bool occ = !((inv_z > invD1w - OCC_THRES_) && inview);

  // Per-pixel warping Jacobian: warped2 = R*(px,py,invD0) + t
  const float jx = R00 * px + R01 * py + R02 * id0 + t0;
  const float jy = R10 * px + R11 * py + R12 * id0 + t1;
  const float jz = R20 * px + R21 * py + R22 * id0 + t2;
  const float xy = jx * jy;
  float Jx[6], Jy[6];
  Jx[0] = -xy * fx;            Jx[1] = (1.0f + jx * jx) * fx;  Jx[2] = -jy * fx;
  Jx[3] = jz * fx;             Jx[4] = 0.0f;                   Jx[5] = -jz * jx * fx;
  Jy[0] = (-1.0f - jy * jy) * fy; Jy[1] = xy * fy;             Jy[2] = jx * fy;
  Jy[3] = 0.0f;                Jy[4] = jz * fy;                Jy[5] = -jz * jy * fy;

  const float* F1b = F1 + (long)b * CHW_;
  const float* GXb = ws + WS_GX + (long)b * CHW_;
  const float* GYb = ws + WS_GY + (long)b * CHW_;
  const unsigned long long F0base =
      (unsigned long long)(const void*)(F0 + (long)b * CHW_);

  v8f acc = {0.0f, 0.0f, 0.0f, 0.0f, 0.0f, 0.0f, 0.0f, 0.0f};

  // Prologue: async-load channel 0's F0 tile into buffer 0.
  async_load_f0(F0base, (unsigned)(p * 4), (unsigned)(size_t)&f0buf[0][tid]);

  for (int c = 0; c < C_; ++c) {
    const long co = (long)c * HW_;
    if (c + 1 < C_) __builtin_prefetch(F1b + co + HW_ + o00, 0, 1);  // global_prefetch_b8

    const float f1w = F1b[co + o00] * w00 + F1b[co + o01] * w01 +
                      F1b[co + o10] * w10 + F1b[co + o11] * w11;
    const float gxw = GXb[co + o00] * w00 + GXb[co + o01] * w01 +
                      GXb[co + o10] * w10 + GXb[co + o11] * w11;
    const float gyw = GYb[co + o00] * w00 + GYb[co + o01] * w01 +
                      GYb[co + o10] * w10 + GYb[co + o11] * w11;

    // Kick off next channel's F0 tile, then wait for the current one.
    if (c + 1 < C_) {
      async_load_f0(F0base, (unsigned)(((c + 1) * HW_ + p) * 4),
                    (unsigned)(size_t)&f0buf[(c + 1) & 1][tid]);
      asm volatile("s_wait_asynccnt 0x1" ::: "memory");  // in-order completion
    } else {
      asm volatile("s_wait_asynccnt 0x0" ::: "memory");
    }
    const float f0v = f0buf[c & 1][tid];

    const float res = occ ? 0.001f : (f1w - f0v);
    const float wgt = 1.0f / (1.0f + res * res);
    if (write_weights) out[(long)b * OUT_STRIDE + 12 + co + p] = wgt;

    // Extended row e = (J0..J5, w*r, 0): EᵀE holds JᵀJ and rhs simultaneously.
    float* row = rowbuf[wave][lane];
#pragma unroll
    for (int k = 0; k < 6; ++k) row[k] = -(gxw * Jx[k] + gyw * Jy[k]);
    row[6] = wgt * res;
    row[7] = 0.0f;
    __syncthreads();

    const float(*rb)[8] = (const float(*)[8])rowbuf[wave];
#pragma unroll
    for (int cb = 0; cb < 32; cb += 4) {
      // f32 16x16x4 fragment: lane l holds e_{cb+2*hlf+k}[col]; A and B
      // per-lane fragments are identical for the EᵀE product.
      v2f a;
      a.x = (col < 8) ? rb[cb + 2 * hlf + 0][col] : 0.0f;
      a.y = (col < 8) ? rb[cb + 2 * hlf + 1][col] : 0.0f;
      acc = __builtin_amdgcn_wmma_f32_16x16x4_f32(false, a, false, a,
                                                  (short)0, acc, false, false);
    }
    __syncthreads();
  }

  // D[M][N]: lane l, vgpr i -> M = i + 8*(l>>4), N = l&15. Need 7x7 block.
  float* part = ws + WS_PART + b * 64;
  if (lane < 7) {
    atomicAdd(&part[0 * 8 + lane], acc[0]);
    atomicAdd(&part[1 * 8 + lane], acc[1]);
    atomicAdd(&part[2 * 8 + lane], acc[2]);
    atomicAdd(&part[3 * 8 + lane], acc[3]);
    atomicAdd(&part[4 * 8 + lane], acc[4]);
    atomicAdd(&part[5 * 8 + lane], acc[5]);
    atomicAdd(&part[6 * 8 + lane], acc[6]);
  }
}

// ---------------------------------------------------------------------------
// Per-batch: damped 6x6 solve, se3_exp, pose update; zero partials for next
// iteration; optionally emit R,t to output.
// ---------------------------------------------------------------------------
__global__ void solve_update_kernel(float* __restrict__ ws, float* __restrict__ out,
                                    int write_out) {
  int b = threadIdx.x;
  if (b >= B_) return;
  float* part = ws + WS_PART + b * 64;

  float M[6][7];
  for (int i = 0; i < 6; ++i) {
    for (int j = 0; j < 6; ++j) M[i][j] = part[i * 8 + j];
    M[i][6] = part[i * 8 + 6];  // rhs = column 6 of EᵀE
    M[i][i] += DAMP_;
  }
  // Gaussian elimination with partial pivoting
  for (int k = 0; k < 6; ++k) {
    int piv = k;
    float best = fabsf(M[k][k]);
    for (int r = k + 1; r < 6; ++r) {
      float vv = fabsf(M[r][k]);
      if (vv > best) { best = vv; piv = r; }
    }
    if (piv != k)
      for (int j = k; j < 7; ++j) {
        float tmp = M[k][j]; M[k][j] = M[piv][j]; M[piv][j] = tmp;
      }
    float inv = 1.0f / M[k][k];
    for (int r = k + 1; r < 6; ++r) {
      float f = M[r][k] * inv;
      for (int j = k; j < 7; ++j) M[r][j] -= f * M[k][j];
    }
  }
  float dp[6];
  for (int i = 5; i >= 0; --i) {
    float s = M[i][6];
    for (int j = i + 1; j < 6; ++j) s -= M[i][j] * dp[j];
    dp[i] = s / M[i][i];
  }

  // se3_exp
  const float w0 = dp[0], w1 = dp[1], w2 = dp[2];
  const float v0 = dp[3], v1 = dp[4], v2 = dp[5];
  const float th2 = w0 * w0 + w1 * w1 + w2 * w2;
  const float th = sqrtf(th2 + 1e-12f);
  const float a = sinf(th) / th;
  const float bb = (1.0f - cosf(th)) / (th2 + 1e-12f);
  const float cc = (1.0f - a) / (th2 + 1e-12f);
  const float Wm[3][3] = {{0.0f, -w2, w1}, {w2, 0.0f, -w0}, {-w1, w0, 0.0f}};
  float W2[3][3];
  for (int i = 0; i < 3; ++i)
    for (int j = 0; j < 3; ++j)
      W2[i][j] = Wm[i][0] * Wm[0][j] + Wm[i][1] * Wm[1][j] + Wm[i][2] * Wm[2][j];
  float dR[3][3], V[3][3];
  for (int i = 0; i < 3; ++i)
    for (int j = 0; j < 3; ++j) {
      float I = (i == j) ? 1.0f : 0.0f;
      dR[i][j] = I + a * Wm[i][j] + bb * W2[i][j];
      V[i][j] = I + bb * Wm[i][j] + cc * W2[i][j];
    }
  const float dt0 = V[0][0] * v0 + V[0][1] * v1 + V[0][2] * v2;
  const float dt1 = V[1][0] * v0 + V[1][1] * v1 + V[1][2] * v2;
  const float dt2 = V[2][0] * v0 + V[2][1] * v1 + V[2][2] * v2;

  float* Rt = ws + WS_RT + b * 12;
  float Ro[3][3] = {{Rt[0], Rt[1], Rt[2]}, {Rt[3], Rt[4], Rt[5]}, {Rt[6], Rt[7], Rt[8]}};
  const float to0 = Rt[9], to1 = Rt[10], to2 = Rt[11];
  float Rn[3][3], tn[3];
  for (int i = 0; i < 3; ++i) {
    for (int j = 0; j < 3; ++j)
      Rn[i][j] = dR[i][0] * Ro[0][j] + dR[i][1] * Ro[1][j] + dR[i][2] * Ro[2][j];
    tn[i] = dR[i][0] * to0 + dR[i][1] * to1 + dR[i][2] * to2;
  }
  tn[0] += dt0; tn[1] += dt1; tn[2] += dt2;
  for (int i = 0; i < 3; ++i)
    for (int j = 0; j < 3; ++j) Rt[i * 3 + j] = Rn[i][j];
  Rt[9] = tn[0]; Rt[10] = tn[1]; Rt[11] = tn[2];

  for (int k = 0; k < 64; ++k) part[k] = 0.0f;  // reset for next iteration

  if (write_out) {
    float* ob = out + (long)b * OUT_STRIDE;
    for (int i = 0; i < 3; ++i)
      for (int j = 0; j < 3; ++j) ob[i * 3 + j] = Rn[i][j];
    ob[9] = tn[0]; ob[10] = tn[1]; ob[11] = tn[2];
  }
}

// ---------------------------------------------------------------------------
extern "C" void kernel_launch(void* const* d_in, const int* in_sizes, int n_in,
                              void* d_out, int out_size, void* d_ws, size_t ws_size,
                              hipStream_t stream) {
  const float* Rin = (const float*)d_in[0];
  const float* tin = (const float*)d_in[1];
  const float* F0 = (const float*)d_in[2];
  const float* F1 = (const float*)d_in[3];
  const float* invD0 = (const float*)d_in[4];
  const float* invD1 = (const float*)d_in[5];
  const float* Kin = (const float*)d_in[6];
  // d_in[7], d_in[8] = sigma0/sigma1 (unused by reference forward)
  float* out = (float*)d_out;
  float* ws = (float*)d_ws;

  init_state_kernel<<<1, 512, 0, stream>>>(Rin, tin, ws);
  sobel_kernel<<<(B_ * CHW_) / 256, 256, 0, stream>>>(F1, ws);

  for (int it = 0; it < MAX_ITER_; ++it) {
    const int last = (it == MAX_ITER_ - 1) ? 1 : 0;
    irls_kernel<<<B_ * (HW_ / 256), 256, 0, stream>>>(F0, F1, invD0, invD1, Kin,
                                                      ws, out, last);
    solve_update_kernel<<<1, 64, 0, stream>>>(ws, out, last);
  }
}